// SPAIRPointFeatureNetwork_15470472200206
// MI455X (gfx1250) — compile-verified
//
#include <hip/hip_runtime.h>
#include <math.h>
#include <stdint.h>

#define N_POINTS 16384
#define MAX_NBR  64
#define RADIUS2  (1.0f / 256.0f)   // (1/16)^2

typedef __attribute__((ext_vector_type(2))) float v2f;
typedef __attribute__((ext_vector_type(8))) float v8f;

__device__ __forceinline__ v8f wmma_f32_16x16x4(v2f a, v2f b, v8f c) {
    // 8 args: (neg_a, A, neg_b, B, c_mod, C, reuse_a, reuse_b)
    return __builtin_amdgcn_wmma_f32_16x16x4_f32(false, a, false, b, (short)0, c,
                                                 false, false);
}

// ---- CDNA5 async global->LDS copies (ASYNCcnt-tracked), GV addressing -------
__device__ __forceinline__ void async_g2l_b32(unsigned lds, const void* g) {
    asm volatile("global_load_async_to_lds_b32 %0, %1, off"
                 :: "v"(lds), "v"((unsigned long long)(uintptr_t)g) : "memory");
}
__device__ __forceinline__ void async_g2l_b64(unsigned lds, const void* g) {
    asm volatile("global_load_async_to_lds_b64 %0, %1, off"
                 :: "v"(lds), "v"((unsigned long long)(uintptr_t)g) : "memory");
}
__device__ __forceinline__ void async_g2l_b128(unsigned lds, const void* g) {
    asm volatile("global_load_async_to_lds_b128 %0, %1, off"
                 :: "v"(lds), "v"((unsigned long long)(uintptr_t)g) : "memory");
}
#define WAIT_ASYNC_4() asm volatile("s_wait_asynccnt 0x4" ::: "memory")
#define WAIT_ASYNC_0() asm volatile("s_wait_asynccnt 0x0" ::: "memory")

// LDS offset of a __shared__ object (generic ptr low 32 bits = wave-relative LDS addr)
__device__ __forceinline__ unsigned lds_off(const void* p) {
    return (unsigned)(uintptr_t)p;
}

// ---------------------------------------------------------------- squared norms
__global__ void k_sq(const float* __restrict__ pos, float* __restrict__ sq) {
    int i = blockIdx.x * blockDim.x + threadIdx.x;
    if (i < N_POINTS) {
        float x = pos[3 * i], y = pos[3 * i + 1], z = pos[3 * i + 2];
        sq[i] = x * x + y * y + z * z;
    }
}

// ---------------------------------------------------------------- radius graph
// Block = 256 threads (8 waves) owns 16 query rows. Each wave sweeps 1/8 of the
// 1024 candidate tiles. Candidate records (x,y,z,|p|^2,batch) are double-buffered
// into LDS with async loads; Gram tile via V_WMMA_F32_16X16X4_F32.
__global__ __launch_bounds__(256) void k_radius(const float* __restrict__ pos,
                                                const float* __restrict__ sq,
                                                const int* __restrict__ batch,
                                                unsigned* __restrict__ cntg,
                                                int* __restrict__ idxg) {
    __shared__ unsigned cnt[16];
    __shared__ int      lists[16][MAX_NBR];
    __shared__ __align__(16) float stage[8][2][16 * 8]; // per point: x y z sq batch pad pad pad

    const int qbase = blockIdx.x * 16;
    const int tid   = threadIdx.x;
    const int lane  = tid & 31;
    const int wid   = tid >> 5;

    if (tid < 16) cnt[tid] = 0u;
    for (int s = tid; s < 16 * MAX_NBR; s += 256)
        lists[s >> 6][s & 63] = qbase + (s >> 6);   // pad with self (valid gather)
    __syncthreads();

    // per-lane row data in registers (half-wave h touches rows h*8 .. h*8+7 only)
    const int rofs = (lane >= 16) ? 8 : 0;
    float rowsq[8];
    int   rowbat[8];
#pragma unroll
    for (int e = 0; e < 8; ++e) {
        rowsq[e]  = sq[qbase + rofs + e];
        rowbat[e] = batch[qbase + rofs + e];
    }

    // A tile (16 queries x K=4), branchless lane split
    v2f a;
    {
        const float* p = pos + 3 * (qbase + (lane & 15));
        float x = p[0], y = p[1], z = p[2];
        a.x = (lane < 16) ? x : z;
        a.y = (lane < 16) ? y : 0.0f;
    }

    const int col   = lane & 15;
    const int TILES = N_POINTS / 16;

    auto prefetch = [&](int ct, int buf) {
        if (lane < 16) {
            const int    j = ct * 16 + lane;
            const float* p = pos + 3 * j;
            unsigned     l = lds_off(&stage[wid][buf][lane * 8]);
            async_g2l_b64(l,      p);          // x,y
            async_g2l_b32(l + 8,  p + 2);      // z
            async_g2l_b32(l + 12, sq + j);     // |p|^2
            async_g2l_b32(l + 16, batch + j);  // batch id (raw bits)
        }
    };

    prefetch(wid, 0);
    int it = 0;
    for (int ct = wid; ct < TILES; ct += 8, ++it) {
        const int nct = ct + 8;
        prefetch(nct < TILES ? nct : ct, (it + 1) & 1);
        WAIT_ASYNC_4();                         // current buffer's 4 loads complete

        const float* sp  = &stage[wid][it & 1][col * 8];
        const float4 rec = *reinterpret_cast<const float4*>(sp);          // ds_load_b128
        const int    cbat = *reinterpret_cast<const int*>(sp + 4);        // ds_load_b32

        v2f b;
        b.x = (lane < 16) ? rec.x : rec.z;
        b.y = (lane < 16) ? rec.y : 0.0f;

        v8f acc = {};
        acc = wmma_f32_16x16x4(a, b, acc);

        const int j = ct * 16 + col;
#pragma unroll
        for (int e = 0; e < 8; ++e) {
            const float d2 = rowsq[e] + rec.w - 2.0f * acc[e];
            if (d2 <= RADIUS2 && rowbat[e] == cbat) {
                unsigned slot = atomicAdd(&cnt[rofs + e], 1u);   // ds_add_rtn_u32
                if (slot < MAX_NBR) lists[rofs + e][slot] = j;
            }
        }
    }
    __syncthreads();

    if (tid < 16) {
        unsigned c = cnt[tid];
        cntg[qbase + tid] = (c > MAX_NBR) ? MAX_NBR : c;
    }
    for (int s = tid; s < 16 * MAX_NBR; s += 256)
        idxg[(qbase + (s >> 6)) * MAX_NBR + (s & 63)] = lists[s >> 6][s & 63];
}

// ---------------------------------------------------------------- PointNetConv
// One wave per point. msg = relu([x_j, pos_j-pos_i] @ W1 + b1), masked max over
// <=64 neighbors, then (agg @ W2 + b2) with celu. First linear on f32 WMMA:
// 4 row tiles (64 nbrs) x KT k-tiles x CT col tiles. Neighbor feature gather is
// streamed to LDS with async loads (b128 bursts when C_IN is a multiple of 4).
template <int C_IN, int C_MID, int C_OUT, int KT, int CT>
__global__ __launch_bounds__(256) void k_conv(const float* __restrict__ xin,
                                              const float* __restrict__ pos,
                                              const unsigned* __restrict__ cntg,
                                              const int* __restrict__ idxg,
                                              const float* __restrict__ W1,
                                              const float* __restrict__ b1,
                                              const float* __restrict__ W2,
                                              const float* __restrict__ b2,
                                              float* __restrict__ xout) {
    constexpr int KPAD = KT * 4;
    constexpr int KDIM = C_IN + 3;
    static_assert((KPAD * 4) % 16 == 0, "row stride must keep 16B alignment");
    __shared__ __align__(16) float feat[8][MAX_NBR][KPAD];
    __shared__ float agg[8][CT * 16];

    const int tid  = threadIdx.x;
    const int lane = tid & 31;
    const int wid  = tid >> 5;
    const int i    = blockIdx.x * 8 + wid;   // grid = N/8 -> always in range

    const unsigned cnt = cntg[i];
    const float pix = pos[3 * i], piy = pos[3 * i + 1], piz = pos[3 * i + 2];

    // stage neighbor features: x_j via async global->LDS, rel-pos + pad via VALU
#pragma unroll
    for (int t = 0; t < 2; ++t) {
        const int s = lane + 32 * t;
        const int j = idxg[i * MAX_NBR + s];
        const unsigned lbase = lds_off(&feat[wid][s][0]);
        if constexpr (C_IN % 4 == 0) {
#pragma unroll
            for (int k = 0; k < C_IN; k += 4)
                async_g2l_b128(lbase + 4u * k, xin + (size_t)j * C_IN + k);
        } else {
#pragma unroll
            for (int k = 0; k < C_IN; ++k)
                async_g2l_b32(lbase + 4u * k, xin + (size_t)j * C_IN + k);
        }
        feat[wid][s][C_IN + 0] = pos[3 * j + 0] - pix;
        feat[wid][s][C_IN + 1] = pos[3 * j + 1] - piy;
        feat[wid][s][C_IN + 2] = pos[3 * j + 2] - piz;
#pragma unroll
        for (int k = KDIM; k < KPAD; ++k) feat[wid][s][k] = 0.0f;
    }
    WAIT_ASYNC_0();
    __syncthreads();

    const int col = lane & 15;

    // B tiles = W1 (KDIM x C_MID, row-major), zero-padded
    v2f   Bt[KT][CT];
    float bias[CT];
#pragma unroll
    for (int kt = 0; kt < KT; ++kt) {
#pragma unroll
        for (int ct = 0; ct < CT; ++ct) {
            const int n  = ct * 16 + col;
            const int k0 = kt * 4 + ((lane >= 16) ? 2 : 0);
            Bt[kt][ct].x = (n < C_MID && k0     < KDIM) ? W1[k0 * C_MID + n]       : 0.0f;
            Bt[kt][ct].y = (n < C_MID && k0 + 1 < KDIM) ? W1[(k0 + 1) * C_MID + n] : 0.0f;
        }
    }
#pragma unroll
    for (int ct = 0; ct < CT; ++ct) {
        const int n = ct * 16 + col;
        bias[ct] = (n < C_MID) ? b1[n] : 0.0f;
    }

    float colmax[CT];
#pragma unroll
    for (int ct = 0; ct < CT; ++ct) colmax[ct] = -3.0e38f;

#pragma unroll
    for (int rt = 0; rt < 4; ++rt) {
        v2f At[KT];
        const int m     = rt * 16 + (lane & 15);
        const int kbase = (lane >= 16) ? 2 : 0;
#pragma unroll
        for (int kt = 0; kt < KT; ++kt) {
            At[kt].x = feat[wid][m][kt * 4 + kbase];
            At[kt].y = feat[wid][m][kt * 4 + kbase + 1];
        }
#pragma unroll
        for (int ct = 0; ct < CT; ++ct) {
            v8f acc = {};
#pragma unroll
            for (int kt = 0; kt < KT; ++kt)
                acc = wmma_f32_16x16x4(At[kt], Bt[kt][ct], acc);
#pragma unroll
            for (int e = 0; e < 8; ++e) {
                const int mrow = rt * 16 + e + ((lane >= 16) ? 8 : 0);
                float h = fmaxf(acc[e] + bias[ct], 0.0f);           // relu
                h = (mrow < (int)cnt) ? h : -3.0e38f;               // neighbor mask
                colmax[ct] = fmaxf(colmax[ct], h);
            }
        }
    }
    // lanes L and L^16 cover the same columns -> one xor-16 max finishes the reduce
#pragma unroll
    for (int ct = 0; ct < CT; ++ct)
        colmax[ct] = fmaxf(colmax[ct], __shfl_xor(colmax[ct], 16, 32));

#pragma unroll
    for (int ct = 0; ct < CT; ++ct) {
        const int n = ct * 16 + col;
        if (lane < 16 && n < C_MID) agg[wid][n] = colmax[ct];
    }
    __syncthreads();

    // second linear + celu (alpha = 1)
    if (lane < C_OUT) {
        float acc = b2[lane];
#pragma unroll
        for (int k = 0; k < C_MID; ++k) acc += agg[wid][k] * W2[k * C_OUT + lane];
        xout[i * C_OUT + lane] = (acc > 0.0f) ? acc : expm1f(acc);
    }
}

// ---------------------------------------------------------------- output tuple
__global__ void k_finalize(const float* __restrict__ pos,
                           const int* __restrict__ batch,
                           float* __restrict__ out) {
    int i = blockIdx.x * blockDim.x + threadIdx.x;
    if (i < N_POINTS) {
        out[3 * i + 0] = pos[3 * i + 0];
        out[3 * i + 1] = pos[3 * i + 1];
        out[3 * i + 2] = pos[3 * i + 2];
        out[N_POINTS * 3 + N_POINTS * 32 + i] = (float)batch[i];
    }
}

extern "C" void kernel_launch(void* const* d_in, const int* in_sizes, int n_in,
                              void* d_out, int out_size, void* d_ws, size_t ws_size,
                              hipStream_t stream) {
    const float* pos   = (const float*)d_in[0];
    // d_in[1] = rgb (accepted but unused by the reference)
    const int*   batch = (const int*)d_in[2];
    const float* W1a = (const float*)d_in[3],  *b1a = (const float*)d_in[4];
    const float* W2a = (const float*)d_in[5],  *b2a = (const float*)d_in[6];
    const float* W1b = (const float*)d_in[7],  *b1b = (const float*)d_in[8];
    const float* W2b = (const float*)d_in[9],  *b2b = (const float*)d_in[10];
    const float* W1c = (const float*)d_in[11], *b1c = (const float*)d_in[12];
    const float* W2c = (const float*)d_in[13], *b2c = (const float*)d_in[14];

    float* out = (float*)d_out;

    // workspace layout
    char*     ws  = (char*)d_ws;
    float*    sq  = (float*)ws;                                   // 64 KB
    unsigned* cnt = (unsigned*)(ws + 64 * 1024);                  // 64 KB
    int*      idx = (int*)(ws + 128 * 1024);                      // 4 MB
    float*    fA  = (float*)(ws + 128 * 1024 + (size_t)N_POINTS * MAX_NBR * 4); // N*8
    float*    fB  = fA + (size_t)N_POINTS * 8;                    // N*16
    float*    outSeg = out + 3 * N_POINTS;                        // N*32 output features

    k_sq<<<N_POINTS / 256, 256, 0, stream>>>(pos, sq);
    k_radius<<<N_POINTS / 16, 256, 0, stream>>>(pos, sq, batch, cnt, idx);

    // layer a: K=6 -> KT=2, C_MID=8  -> CT=1
    k_conv<3, 8, 8, 2, 1><<<N_POINTS / 8, 256, 0, stream>>>(
        pos, pos, cnt, idx, W1a, b1a, W2a, b2a, fA);
    // layer b: K=11 -> KT=3, C_MID=16 -> CT=1
    k_conv<8, 16, 16, 3, 1><<<N_POINTS / 8, 256, 0, stream>>>(
        fA, pos, cnt, idx, W1b, b1b, W2b, b2b, fB);
    // layer c: K=19 -> KT=5, C_MID=32 -> CT=2, writes straight into output tuple
    k_conv<16, 32, 32, 5, 2><<<N_POINTS / 8, 256, 0, stream>>>(
        fB, pos, cnt, idx, W1c, b1c, W2c, b2c, outSeg);

    k_finalize<<<N_POINTS / 256, 256, 0, stream>>>(pos, batch, out);
}